// IntraPartMR_6356551598737
// MI455X (gfx1250) — compile-verified
//
#include <hip/hip_runtime.h>

typedef __attribute__((ext_vector_type(2))) float v2f;
typedef __attribute__((ext_vector_type(8))) float v8f;

#define B_    32
#define C_    128
#define P_    4096
#define C2_   256
#define GS    64
#define BNEPS 1e-5f

// workspace layout (floats):
//   [WS_BIAS,  +8192)    bias_eff[b][oc]
//   [WS_PART,  +131072)  per-stats-block partials: 128 ch x {sum,sumsq}
//   [WS_SCALE, +256)     per-channel scale
//   [WS_SHIFT, +256)     per-channel shift
#define WS_BIAS  0
#define WS_PART  8192
#define WS_SCALE (8192 + 512 * 256)
#define WS_SHIFT (WS_SCALE + 256)

#if __has_builtin(__builtin_amdgcn_global_load_async_to_lds_b128)
#define USE_ASYNC_LDS 1
#else
#define USE_ASYNC_LDS 0
#endif

typedef int v4i_raw __attribute__((vector_size(16)));
typedef __attribute__((address_space(1))) v4i_raw* gv4i_ptr;  // global int4*
typedef __attribute__((address_space(3))) v4i_raw* lv4i_ptr;  // LDS int4*

__device__ __forceinline__ void async_wait0() {
#if __has_builtin(__builtin_amdgcn_s_wait_asynccnt)
  __builtin_amdgcn_s_wait_asynccnt(0);
#else
  asm volatile("s_wait_asynccnt 0x0" ::: "memory");
#endif
}

// ---------------- Kernel 1: per-(b,c) window max + effective bias ----------
__global__ __launch_bounds__(256) void prep_kernel(const float* __restrict__ x,
                                                   const float* __restrict__ W,
                                                   const float* __restrict__ bias,
                                                   float* __restrict__ ws) {
  __shared__ float m[C_];
  int b = blockIdx.x;
  int t = threadIdx.x;  // 256
  if (t < C_) {
    const float* xp = x + ((size_t)(b * C_ + t)) * P_;
    m[t] = fmaxf(xp[1], fmaxf(xp[2], xp[3]));
  }
  __syncthreads();
  int oc = t;
  float be = bias[oc];
  if (oc >= 128) {
    int cblk = (oc >> 6) - 2;  // 0 or 1
    const float* wr = W + oc * GS;
    const float* mp = m + cblk * GS;
    float s = 0.f;
#pragma unroll 8
    for (int i = 0; i < GS; ++i) s += wr[i] * mp[i];
    be += s;
  }
  ws[WS_BIAS + b * C2_ + oc] = be;
}

// ---------------- Kernel 2/4: WMMA GEMM core (stats or output epilogue) ----
template <int PT_ITERS, bool STATS>
__global__ __launch_bounds__(256) void fused_gemm(const float* __restrict__ x,
                                                  const float* __restrict__ W,
                                                  float* __restrict__ ws,
                                                  float* __restrict__ out) {
  constexpr int NCHUNK = 64 / PT_ITERS;  // chunks of (PT_ITERS*64) points
  int blk   = blockIdx.x;
  int chunk = blk % NCHUNK;
  int cblk  = (blk / NCHUNK) & 1;
  int b     = blk / (NCHUNK * 2);

  int tid  = threadIdx.x;
  int wave = tid >> 5;   // 0..7
  int lane = tid & 31;
  int half = lane >> 4;  // 0/1
  int lm   = lane & 15;

  int   g_half = wave >> 2;        // 0: g=cblk, 1: g=cblk+2
  int   o0     = (wave & 3) << 4;  // 0,16,32,48
  int   g      = cblk + 2 * g_half;
  float sign   = g_half ? -1.f : 1.f;  // fold (m - x) into -W, bias_eff
  int   cbase  = cblk * GS;

  __shared__ float ldsx[GS * 64];  // 64 ch x 64 pts (16 KB)

  // A fragments in registers for the whole kernel: a[s] covers K = 4s..4s+3
  v2f afrag[16];
  {
    const float* wr = W + (size_t)(g * GS + o0 + lm) * GS;
#pragma unroll
    for (int s = 0; s < 16; ++s) {
      int k = s * 4 + 2 * half;
      v2f a;
      a.x = sign * wr[k];
      a.y = sign * wr[k + 1];
      afrag[s] = a;
    }
  }

  float biasv[8];
#pragma unroll
  for (int v = 0; v < 8; ++v)
    biasv[v] = ws[WS_BIAS + b * C2_ + g * GS + o0 + v + 8 * half];

  float scalev[8], shiftv[8];
  if (!STATS) {
#pragma unroll
    for (int v = 0; v < 8; ++v) {
      int oc    = g * GS + o0 + v + 8 * half;
      scalev[v] = ws[WS_SCALE + oc];
      shiftv[v] = ws[WS_SHIFT + oc];
    }
  }

  float psum[8], psumsq[8];
#pragma unroll
  for (int v = 0; v < 8; ++v) { psum[v] = 0.f; psumsq[v] = 0.f; }

  const v8f zerov = {0.f, 0.f, 0.f, 0.f, 0.f, 0.f, 0.f, 0.f};

  for (int pt = 0; pt < PT_ITERS; ++pt) {
    int p0 = chunk * (PT_ITERS * 64) + pt * 64;

    __syncthreads();  // protect previous iteration's LDS reads
#pragma unroll
    for (int r = 0; r < 4; ++r) {
      int idx = r * 256 + tid;  // 0..1023
      int row = idx >> 4;
      int q   = idx & 15;
      const float* src = x + ((size_t)(b * C_ + cbase + row)) * P_ + p0 + q * 4;
      float*       dst = ldsx + row * 64 + q * 4;
#if USE_ASYNC_LDS
      __builtin_amdgcn_global_load_async_to_lds_b128((gv4i_ptr)src, (lv4i_ptr)dst, 0, 0);
#else
      *(float4*)dst = *(const float4*)src;
#endif
    }
#if USE_ASYNC_LDS
    async_wait0();
#endif
    __syncthreads();

    v8f acc[4];
#pragma unroll
    for (int n = 0; n < 4; ++n) acc[n] = zerov;

#pragma unroll
    for (int s = 0; s < 16; ++s) {
      int krow = s * 4 + 2 * half;
      v2f a    = afrag[s];
#pragma unroll
      for (int n = 0; n < 4; ++n) {
        v2f bf;
        bf.x = ldsx[krow * 64 + n * 16 + lm];
        bf.y = ldsx[(krow + 1) * 64 + n * 16 + lm];
        acc[n] = __builtin_amdgcn_wmma_f32_16x16x4_f32(
            false, a, false, bf, (short)0, acc[n], false, false);
      }
    }

    if (STATS) {
#pragma unroll
      for (int n = 0; n < 4; ++n)
#pragma unroll
        for (int v = 0; v < 8; ++v) {
          float y = acc[n][v] + biasv[v];
          psum[v] += y;
          psumsq[v] += y * y;
        }
    } else {
#pragma unroll
      for (int n = 0; n < 4; ++n) {
        int p = p0 + n * 16 + lm;
#pragma unroll
        for (int v = 0; v < 8; ++v) {
          float y  = acc[n][v] + biasv[v];
          float rr = fmaxf(y * scalev[v] + shiftv[v], 0.f);
          int   oc = g * GS + o0 + v + 8 * half;
          out[((size_t)(b * C2_ + oc)) * P_ + p] = rr;
        }
      }
    }
  }

  if (STATS) {
    // reduce within each 16-lane half (xor 8,4,2,1 stays inside the half)
#pragma unroll
    for (int v = 0; v < 8; ++v) {
      float s  = psum[v];
      float ss = psumsq[v];
#pragma unroll
      for (int off = 8; off >= 1; off >>= 1) {
        s  += __shfl_xor(s, off, 32);
        ss += __shfl_xor(ss, off, 32);
      }
      if (lm == 0) {
        int    lc   = g_half * 64 + o0 + v + 8 * half;  // 0..127
        float* part = ws + WS_PART + (size_t)blockIdx.x * 256 + lc * 2;
        part[0] = s;
        part[1] = ss;
      }
    }
  }
}

// ---------------- Kernel 3: per-channel BN scale/shift ---------------------
__global__ __launch_bounds__(256) void finalize_kernel(const float* __restrict__ gamma,
                                                       const float* __restrict__ beta,
                                                       float* __restrict__ ws) {
  int oc     = blockIdx.x;       // 256 channels
  int g      = oc >> 6;
  int cblk   = g & 1;
  int g_half = g >> 1;
  int lc     = g_half * 64 + (oc & 63);
  int t      = threadIdx.x;      // 256 = 32 b x 8 chunks
  int b      = t >> 3;
  int chunk  = t & 7;
  int sblk   = (b * 2 + cblk) * 8 + chunk;  // matches fused_gemm<8,true> grid decode

  const float* part = ws + WS_PART + (size_t)sblk * 256 + lc * 2;
  __shared__ float sh[256], shs[256];
  sh[t]  = part[0];
  shs[t] = part[1];
  __syncthreads();
  for (int off = 128; off > 0; off >>= 1) {
    if (t < off) {
      sh[t]  += sh[t + off];
      shs[t] += shs[t + off];
    }
    __syncthreads();
  }
  if (t == 0) {
    float inv  = 1.f / ((float)B_ * (float)P_);
    float mean = sh[0] * inv;
    float var  = shs[0] * inv - mean * mean;
    float sc   = gamma[oc] * rsqrtf(var + BNEPS);
    ws[WS_SCALE + oc] = sc;
    ws[WS_SHIFT + oc] = beta[oc] - mean * sc;
  }
}

// ---------------------------------------------------------------------------
extern "C" void kernel_launch(void* const* d_in, const int* in_sizes, int n_in,
                              void* d_out, int out_size, void* d_ws, size_t ws_size,
                              hipStream_t stream) {
  const float* x     = (const float*)d_in[0];
  const float* W     = (const float*)d_in[1];
  const float* bias  = (const float*)d_in[2];
  const float* gamma = (const float*)d_in[3];
  const float* beta  = (const float*)d_in[4];
  float*       out   = (float*)d_out;
  float*       ws    = (float*)d_ws;

  prep_kernel<<<B_, 256, 0, stream>>>(x, W, bias, ws);
  // stats pass: 32 b x 2 cblk x 8 chunks(512 pts) = 512 blocks
  fused_gemm<8, true><<<B_ * 2 * 8, 256, 0, stream>>>(x, W, ws, out);
  finalize_kernel<<<C2_, 256, 0, stream>>>(gamma, beta, ws);
  // output pass: 32 x 2 x 64 tiles = 4096 blocks (x re-read hits L2: 64 MiB < 192 MB)
  fused_gemm<1, false><<<B_ * 2 * 64, 256, 0, stream>>>(x, W, ws, out);
}